// GraphNet_88175678586950
// MI455X (gfx1250) — compile-verified
//
#include <hip/hip_runtime.h>
#include <math.h>

#define NS 80
#define NT 80
#define DD 512
#define NN (NS * NT)       /* 6400 */
#define NB 64              /* Cholesky panel width */
#define NBLK (NN / NB)     /* 100 */

typedef __attribute__((ext_vector_type(2))) float v2f;
typedef __attribute__((ext_vector_type(8))) float v8f;

/* ---------------- prologue: graph message passing (tiny, VALU) ------------- */

// Mp0[i][j] = e1[i]·e2[j] + iou[i][j];  e1[i][d] = Us[d*NS+i]
__global__ void k_mp0(const float* __restrict__ Us, const float* __restrict__ Ut,
                      const float* __restrict__ iou, float* __restrict__ Mp0) {
  int i = blockIdx.x, j = threadIdx.x;
  float s = 0.f;
  for (int d = 0; d < DD; ++d) s += Us[d * NS + i] * Ut[d * NT + j];
  Mp0[i * NT + j] = s + iou[i * NT + j];
}

// m1 = Mp0 @ e2 ; m2 = Mp0.T @ e1
__global__ void k_msg(const float* __restrict__ Us, const float* __restrict__ Ut,
                      const float* __restrict__ Mp0,
                      float* __restrict__ m1, float* __restrict__ m2) {
  int d = threadIdx.x, i = blockIdx.x;
  if (blockIdx.y == 0) {
    float s = 0.f;
    for (int j = 0; j < NT; ++j) s += Mp0[i * NT + j] * Ut[d * NT + j];
    m1[i * DD + d] = s;
  } else {
    float s = 0.f;
    for (int j = 0; j < NS; ++j) s += Mp0[j * NT + i] * Us[d * NS + j];
    m2[i * DD + d] = s;
  }
}

__global__ void k_lam(const float* __restrict__ Us, const float* __restrict__ Ut,
                      const float* __restrict__ m1, const float* __restrict__ m2,
                      float* __restrict__ lam1, float* __restrict__ lam2) {
  int i = threadIdx.x;  // 0..159
  if (i < NS) {
    float ne = 0.f, nm = 0.f;
    for (int d = 0; d < DD; ++d) {
      float e = Us[d * NS + i]; ne += e * e;
      float m = m1[i * DD + d]; nm += m * m;
    }
    lam1[i] = sqrtf(ne) / fmaxf(sqrtf(nm), 1e-30f);
  } else {
    int t = i - NS;
    float ne = 0.f, nm = 0.f;
    for (int d = 0; d < DD; ++d) {
      float e = Ut[d * NT + t]; ne += e * e;
      float m = m2[t * DD + d]; nm += m * m;
    }
    lam2[t] = sqrtf(ne) / fmaxf(sqrtf(nm), 1e-30f);
  }
}

// h = relu((e + lam*m) @ W.T + b)
__global__ void k_linear(const float* __restrict__ Us, const float* __restrict__ Ut,
                         const float* __restrict__ m1, const float* __restrict__ m2,
                         const float* __restrict__ lam1, const float* __restrict__ lam2,
                         const float* __restrict__ W, const float* __restrict__ b,
                         float* __restrict__ h1, float* __restrict__ h2) {
  int c = threadIdx.x, i = blockIdx.x;
  const float* Wr = W + (size_t)c * DD;
  float s = b[c];
  if (blockIdx.y == 0) {
    float l = lam1[i];
    for (int d = 0; d < DD; ++d) s += (Us[d * NS + i] + l * m1[i * DD + d]) * Wr[d];
    h1[i * DD + c] = fmaxf(s, 0.f);
  } else {
    float l = lam2[i];
    for (int d = 0; d < DD; ++d) s += (Ut[d * NT + i] + l * m2[i * DD + d]) * Wr[d];
    h2[i * DD + c] = fmaxf(s, 0.f);
  }
}

__global__ void k_invnorm(const float* __restrict__ h1, const float* __restrict__ h2,
                          float* __restrict__ invn) {
  int i = threadIdx.x;  // 0..159
  const float* h = (i < NS) ? (h1 + (size_t)i * DD) : (h2 + (size_t)(i - NS) * DD);
  float s = 0.f;
  for (int d = 0; d < DD; ++d) s += h[d] * h[d];
  invn[i] = 1.f / fmaxf(sqrtf(s), 1e-12f);
}

__global__ void k_scale(float* __restrict__ h1, float* __restrict__ h2,
                        const float* __restrict__ invn) {
  int d = threadIdx.x, i = blockIdx.x;
  if (blockIdx.y == 0) h1[i * DD + d] *= invn[i];
  else                 h2[i * DD + d] *= invn[NS + i];
}

// Mp = e1n @ e2n.T (-> out[0:6400]) and thr_flag (-> tail of out)
__global__ void k_mp_flag(const float* __restrict__ e1n, const float* __restrict__ e2n,
                          const float* __restrict__ kf, const float* __restrict__ iou,
                          const float* __restrict__ thr, float* __restrict__ out) {
  int i = blockIdx.x, j = threadIdx.x;
  float s = 0.f;
  for (int d = 0; d < DD; ++d) s += e1n[i * DD + d] * e2n[j * DD + d];
  int ij = i * NT + j;
  out[ij] = s;
  float f = ((kf[ij] == -1.f) || (iou[ij] == 0.f) || (s < thr[0])) ? 1.f : 0.f;
  out[(size_t)NN + (size_t)NN * NN + ij] = f;
}

/* A[p][q] = k*I - M ;  M[a*ns+i][b*ns+j] = (i!=j && a!=b) ? 0.5*(Mp[i,a]+Mp[j,b]) : 0 */
__global__ void k_buildA(const float* __restrict__ Mp, float* __restrict__ A) {
  long long idx = (long long)blockIdx.x * blockDim.x + threadIdx.x;
  if (idx >= (long long)NN * NN) return;
  int p = (int)(idx / NN), q = (int)(idx % NN);
  int i = p % NS, a = p / NS, j = q % NS, bb = q / NS;
  float v = 0.f;
  if ((i != j) && (a != bb)) v = 0.5f * (Mp[i * NT + a] + Mp[j * NT + bb]);
  A[idx] = ((p == q) ? (float)((NS - 1) * (NT - 1)) : 0.f) - v;
}

/* ---------------- blocked right-looking Cholesky (fp32, in place) ---------- */

// Factor diagonal NB x NB block (single workgroup, LDS-resident)
__global__ void k_potf2(float* __restrict__ A, int kb) {
  __shared__ float s[NB * NB];
  const int base = kb * NB;
  int tid = threadIdx.x; const int T = blockDim.x;
  for (int t = tid; t < NB * NB; t += T)
    s[t] = A[(size_t)(base + t / NB) * NN + base + (t % NB)];
  __syncthreads();
  for (int j = 0; j < NB; ++j) {
    if (tid == 0) s[j * NB + j] = sqrtf(s[j * NB + j]);
    __syncthreads();
    float d = s[j * NB + j];
    for (int i = j + 1 + tid; i < NB; i += T) s[i * NB + j] /= d;
    __syncthreads();
    for (int c = j + 1; c < NB; ++c) {
      float l = s[c * NB + j];
      for (int i = c + tid; i < NB; i += T) s[i * NB + c] -= s[i * NB + j] * l;
    }
    __syncthreads();
  }
  for (int t = tid; t < NB * NB; t += T)
    A[(size_t)(base + t / NB) * NN + base + (t % NB)] = s[t];
}

// Panel: A[row][base..base+NB) <- A_row * Lkk^{-T}, one thread per row
__global__ void k_trsm(float* __restrict__ A, int kb) {
  __shared__ float Lkk[NB * NB];
  const int base = kb * NB;
  int tid = threadIdx.x;
  for (int t = tid; t < NB * NB; t += blockDim.x)
    Lkk[t] = A[(size_t)(base + t / NB) * NN + base + (t % NB)];
  __syncthreads();
  int row = base + NB + blockIdx.x * blockDim.x + tid;
  if (row >= NN) return;
  float* a = A + (size_t)row * NN + base;
  float x[NB];
  for (int c = 0; c < NB; ++c) {
    float sum = a[c];
    for (int t = 0; t < c; ++t) sum -= x[t] * Lkk[c * NB + t];
    x[c] = sum / Lkk[c * NB + c];
  }
  for (int c = 0; c < NB; ++c) a[c] = x[c];
}

// Trailing update: one wave computes a 32x32 block (2x2 of 16x16 WMMA tiles),
// C -= P_i @ P_j^T with fp32 WMMA. Register blocking gives 2x fragment reuse:
// 4 WMMAs per 4 fragment loads -> 8 FLOP/B from L2 (A fits in the 192MB L2).
// A-frag (16x4 f32): lane&15 = row, (lane>>4) selects K pair {0,1}/{2,3}.
// B-frag (4x16 f32 = P_j^T): lane&15 = col, same K-pair split.
// C/D (16x16 f32): VGPR r <-> row r + 8*(lane>>4); lane&15 <-> col.
__global__ void __launch_bounds__(32) k_syrk(float* __restrict__ A, int kb) {
  const int off = (kb + 1) * NB;
  const int k0 = kb * NB;
  int ti = blockIdx.x, tj = blockIdx.y;
  if (tj > ti) return;  // lower trailing triangle at 32-block granularity
  const int row0 = off + ti * 32, col0 = off + tj * 32;
  const int lane = threadIdx.x;
  const int half = lane >> 4, lr = lane & 15;
  const float* __restrict__ Ap0 = A + (size_t)(row0 + lr) * NN + k0 + 2 * half;
  const float* __restrict__ Ap1 = Ap0 + (size_t)16 * NN;
  const float* __restrict__ Bp0 = A + (size_t)(col0 + lr) * NN + k0 + 2 * half;
  const float* __restrict__ Bp1 = Bp0 + (size_t)16 * NN;
  v8f acc00 = {}, acc01 = {}, acc10 = {}, acc11 = {};
  for (int t = 0; t < NB; t += 4) {
    v2f a0; a0.x = Ap0[t]; a0.y = Ap0[t + 1];
    v2f a1; a1.x = Ap1[t]; a1.y = Ap1[t + 1];
    v2f b0; b0.x = Bp0[t]; b0.y = Bp0[t + 1];
    v2f b1; b1.x = Bp1[t]; b1.y = Bp1[t + 1];
    acc00 = __builtin_amdgcn_wmma_f32_16x16x4_f32(false, a0, false, b0,
                                                  (short)0, acc00, false, false);
    acc01 = __builtin_amdgcn_wmma_f32_16x16x4_f32(false, a0, false, b1,
                                                  (short)0, acc01, false, false);
    acc10 = __builtin_amdgcn_wmma_f32_16x16x4_f32(false, a1, false, b0,
                                                  (short)0, acc10, false, false);
    acc11 = __builtin_amdgcn_wmma_f32_16x16x4_f32(false, a1, false, b1,
                                                  (short)0, acc11, false, false);
  }
  float* C00 = A + (size_t)row0 * NN + col0;
  float* C01 = C00 + 16;
  float* C10 = C00 + (size_t)16 * NN;
  float* C11 = C10 + 16;
#pragma unroll
  for (int r = 0; r < 8; ++r) {
    size_t ro = (size_t)(r + 8 * half) * NN + lr;
    C00[ro] -= acc00[r];
    C01[ro] -= acc01[r];
    C10[ro] -= acc10[r];
    C11[ro] -= acc11[r];
  }
}

__global__ void k_zero_upper(float* __restrict__ L) {
  long long idx = (long long)blockIdx.x * blockDim.x + threadIdx.x;
  if (idx >= (long long)NN * NN) return;
  int r = (int)(idx / NN), c = (int)(idx % NN);
  if (c > r) L[idx] = 0.f;
}

/* --------------------------------- driver ---------------------------------- */

extern "C" void kernel_launch(void* const* d_in, const int* in_sizes, int n_in,
                              void* d_out, int out_size, void* d_ws, size_t ws_size,
                              hipStream_t stream) {
  (void)in_sizes; (void)n_in; (void)out_size; (void)ws_size;
  const float* Us  = (const float*)d_in[0];   // U_src [1,512,80]
  const float* Ut  = (const float*)d_in[1];   // U_tgt [1,512,80]
  const float* kf  = (const float*)d_in[2];   // kf_gate [80,80]
  const float* thr = (const float*)d_in[3];   // reid_thr [1]
  const float* iou = (const float*)d_in[4];   // iou [80,80]
  const float* W   = (const float*)d_in[5];   // [512,512]
  const float* b   = (const float*)d_in[6];   // [512]

  float* out = (float*)d_out;
  float* Mp  = out;           // [80,80]
  float* L   = out + NN;      // [6400,6400], factored in place

  float* ws   = (float*)d_ws;
  float* Mp0  = ws;                 // 6400
  float* m1   = Mp0 + NN;           // 80*512
  float* m2   = m1 + NS * DD;       // 80*512
  float* lam1 = m2 + NT * DD;       // 80
  float* lam2 = lam1 + NS;          // 80
  float* h1   = lam2 + NT;          // 80*512 -> e1n
  float* h2   = h1 + NS * DD;       // 80*512 -> e2n
  float* invn = h2 + NT * DD;       // 160

  k_mp0<<<NS, NT, 0, stream>>>(Us, Ut, iou, Mp0);
  k_msg<<<dim3(NS, 2), DD, 0, stream>>>(Us, Ut, Mp0, m1, m2);
  k_lam<<<1, NS + NT, 0, stream>>>(Us, Ut, m1, m2, lam1, lam2);
  k_linear<<<dim3(NS, 2), DD, 0, stream>>>(Us, Ut, m1, m2, lam1, lam2, W, b, h1, h2);
  k_invnorm<<<1, NS + NT, 0, stream>>>(h1, h2, invn);
  k_scale<<<dim3(NS, 2), DD, 0, stream>>>(h1, h2, invn);
  k_mp_flag<<<NS, NT, 0, stream>>>(h1, h2, kf, iou, thr, out);

  const int threads = 256;
  const long long total = (long long)NN * NN;
  const int blocksA = (int)((total + threads - 1) / threads);
  k_buildA<<<blocksA, threads, 0, stream>>>(Mp, L);

  for (int kb = 0; kb < NBLK; ++kb) {
    k_potf2<<<1, 256, 0, stream>>>(L, kb);
    int rem = NN - (kb + 1) * NB;
    if (rem > 0) {
      k_trsm<<<(rem + 127) / 128, 128, 0, stream>>>(L, kb);
      int tiles32 = rem / 32;   // rem is always a multiple of 64
      k_syrk<<<dim3(tiles32, tiles32), 32, 0, stream>>>(L, kb);
    }
  }
  k_zero_upper<<<blocksA, threads, 0, stream>>>(L);
}